// PostProcess_2001454760157
// MI455X (gfx1250) — compile-verified
//
#include <hip/hip_runtime.h>
#include <math.h>

#define KSEL 100
#define NMS_TH 0.1f
#define BELONG_TH 0.6f
#define SCORE_TH 0.1f
#define MAXN 16384

typedef float v2f __attribute__((ext_vector_type(2)));
typedef float v8f __attribute__((ext_vector_type(8)));

__device__ __forceinline__ float sigmoidf(float x) {
    return 1.0f / (1.0f + __expf(-x));
}

// ---------------------------------------------------------------------------
// Top-K (K=100) by logit per row, sigmoid applied to selected values.
// One workgroup per image row. The whole row (<= 16384 floats = 64 KB) is
// staged into LDS once via CDNA5 async global->LDS B128 copies (ASYNCcnt),
// then all 100 argmax passes run out of LDS; selected slots are set to -inf.
// ---------------------------------------------------------------------------
__global__ void topk_sigmoid_kernel(const float* __restrict__ logits, int N,
                                    int* __restrict__ out_idx,
                                    float* __restrict__ out_val) {
    const int b = blockIdx.x;
    const float* src = logits + (size_t)b * (size_t)N;
    int* oi = out_idx + b * KSEL;
    float* ov = out_val + b * KSEL;

    __shared__ float sv[MAXN];    // 64 KB staging (gfx1250: 160KB+ addressable)
    __shared__ float wv[8];
    __shared__ int wi[8];

    // --- async stage: each lane moves 16 B global -> LDS (N % 4 == 0) ---
    {
        const unsigned lds_base = (unsigned)(unsigned long long)(&sv[0]);
        for (int k4 = threadIdx.x; k4 < (N >> 2); k4 += blockDim.x) {
            unsigned lds_off = lds_base + (unsigned)(k4 * 16);
            unsigned long long ga = (unsigned long long)(src + k4 * 4);
            asm volatile("global_load_async_to_lds_b128 %0, %1, off"
                         :: "v"(lds_off), "v"(ga) : "memory");
        }
        asm volatile("s_wait_asynccnt 0" ::: "memory");
    }
    __syncthreads();

    const int lane = threadIdx.x & 31;
    const int wid = threadIdx.x >> 5;
    const int nw = blockDim.x >> 5;

    for (int p = 0; p < KSEL; ++p) {
        float bv = -INFINITY;
        int bi = 0x7fffffff;
        // vectorized LDS scan: thread t covers float4 chunks t, t+256, ...
        for (int k = threadIdx.x * 4; k < N; k += blockDim.x * 4) {
            const float4 q = *(const float4*)&sv[k];
            if (q.x > bv || (q.x == bv && k + 0 < bi)) { bv = q.x; bi = k + 0; }
            if (q.y > bv || (q.y == bv && k + 1 < bi)) { bv = q.y; bi = k + 1; }
            if (q.z > bv || (q.z == bv && k + 2 < bi)) { bv = q.z; bi = k + 2; }
            if (q.w > bv || (q.w == bv && k + 3 < bi)) { bv = q.w; bi = k + 3; }
        }
        // wave32 reduce (lower index wins ties, matching lax.top_k)
        for (int off = 16; off > 0; off >>= 1) {
            float ovv = __shfl_down(bv, off, 32);
            int oii = __shfl_down(bi, off, 32);
            if (ovv > bv || (ovv == bv && oii < bi)) { bv = ovv; bi = oii; }
        }
        if (lane == 0) { wv[wid] = bv; wi[wid] = bi; }
        __syncthreads();
        if (threadIdx.x == 0) {
            float fv = wv[0]; int fi = wi[0];
            for (int w2 = 1; w2 < nw; ++w2)
                if (wv[w2] > fv || (wv[w2] == fv && wi[w2] < fi)) {
                    fv = wv[w2]; fi = wi[w2];
                }
            sv[fi] = -INFINITY;   // suppress for later passes
            oi[p] = fi;
            ov[p] = sigmoidf(fv);
        }
        __syncthreads();
    }
}

// ---------------------------------------------------------------------------
// Sequential NMS over 100 candidates resident in LDS. Thread t owns cand t.
// Replicates: sup = (iou[i] > th) & (idx > i) & keep[i]; keep &= ~sup
// ---------------------------------------------------------------------------
__device__ void nms_inplace(float (*box)[4], float* area, int* keepf, int t) {
    if (t < KSEL) {
        float x1 = box[t][0], y1 = box[t][1], x2 = box[t][2], y2 = box[t][3];
        area[t] = (x2 - x1) * (y2 - y1);
    }
    __syncthreads();
    for (int i = 0; i < KSEL; ++i) {
        int ki = keepf[i];
        float ix1 = box[i][0], iy1 = box[i][1];
        float ix2 = box[i][2], iy2 = box[i][3];
        float ia = area[i];
        if (t < KSEL && t > i && ki) {
            float tlx = fmaxf(box[t][0], ix1), tly = fmaxf(box[t][1], iy1);
            float brx = fminf(box[t][2], ix2), bry = fminf(box[t][3], iy2);
            float iw = fmaxf(brx - tlx, 0.f), ih = fmaxf(bry - tly, 0.f);
            float inter = iw * ih;
            float iou = inter / (area[t] + ia - inter);
            if (iou > NMS_TH) keepf[t] = 0;
        }
        __syncthreads();
    }
}

// ---------------------------------------------------------------------------
// Blocks (praw==nullptr: clip, no parent) and Lines (parent belong filter).
// One workgroup (128 threads) per image.
// ---------------------------------------------------------------------------
__global__ void box_filter_kernel(const int* __restrict__ idx_in,
                                  const float* __restrict__ val_in,
                                  const float* __restrict__ raw, int N,
                                  const float* __restrict__ praw, int NP, int ef,
                                  const float* __restrict__ tsizes,
                                  float* __restrict__ out_data,
                                  float* __restrict__ out_scores,
                                  float* __restrict__ out_keep) {
    const int b = blockIdx.x;
    const int t = threadIdx.x;
    const int* sidx = idx_in + b * KSEL;
    const float* vals = val_in + b * KSEL;
    const float imh = tsizes[2 * b + 0], imw = tsizes[2 * b + 1];

    __shared__ float box[KSEL][4];
    __shared__ float area[KSEL];
    __shared__ int keepf[KSEL];
    __shared__ float belong[KSEL];
    __shared__ int s_anyk, s_anyvk, s_fbi;

    float v = 0.f;
    if (t < KSEL) {
        v = vals[t];
        keepf[t] = (v > SCORE_TH) ? 1 : 0;
        int si = sidx[t];
        const float* r = raw + ((size_t)b * N + si) * 4;
        float cx = r[0], cy = r[1], w2 = r[2] * 0.5f, h2 = r[3] * 0.5f;
        float x1 = (cx - w2) * imw, y1 = (cy - h2) * imh;
        float x2 = (cx + w2) * imw, y2 = (cy + h2) * imh;
        if (praw == nullptr) {  // clip only happens in the parentless branch
            x1 = fminf(fmaxf(x1, 0.f), imw); y1 = fminf(fmaxf(y1, 0.f), imh);
            x2 = fminf(fmaxf(x2, 0.f), imw); y2 = fminf(fmaxf(y2, 0.f), imh);
        }
        box[t][0] = x1; box[t][1] = y1; box[t][2] = x2; box[t][3] = y2;
        if (praw != nullptr) {
            int pi = si / ef;
            const float* pr = praw + ((size_t)b * NP + pi) * 4;
            float pcx = pr[0], pcy = pr[1], pw2 = pr[2] * 0.5f, ph2 = pr[3] * 0.5f;
            float px1 = (pcx - pw2) * imw, py1 = (pcy - ph2) * imh;
            float px2 = (pcx + pw2) * imw, py2 = (pcy + ph2) * imh;
            float ix1 = fmaxf(x1, px1), iy1 = fmaxf(y1, py1);
            float ix2 = fminf(x2, px2), iy2 = fminf(y2, py2);
            float inter = fmaxf(ix2 - ix1, 0.f) * fmaxf(iy2 - iy1, 0.f);
            float ca = (x2 - x1) * (y2 - y1);
            belong[t] = inter / (ca + 1e-6f);
        }
    }
    __syncthreads();
    if (t == 0) {
        int a = 0;
        for (int j = 0; j < KSEL; ++j) a |= keepf[j];
        s_anyk = a;
    }
    __syncthreads();
    if (t < KSEL) keepf[t] = s_anyk ? keepf[t] : (t == 0);
    __syncthreads();
    if (praw != nullptr) {
        if (t == 0) {
            int avk = 0; float fv = -INFINITY; int fi = 0;
            for (int j = 0; j < KSEL; ++j) {
                if (keepf[j] && belong[j] > BELONG_TH) avk = 1;
                float s = keepf[j] ? belong[j] : -INFINITY;
                if (s > fv) { fv = s; fi = j; }
            }
            s_anyvk = avk; s_fbi = fi;
        }
        __syncthreads();
        if (t < KSEL) {
            int valid = s_anyvk ? (belong[t] > BELONG_TH) : (t == s_fbi);
            keepf[t] = keepf[t] && valid;
        }
        __syncthreads();
    }
    nms_inplace(box, area, keepf, t);
    if (t < KSEL) {
        int nk = keepf[t];
        out_scores[b * KSEL + t] = nk ? v : 0.f;
        out_keep[b * KSEL + t] = nk ? 1.f : 0.f;
        float* od = out_data + ((size_t)b * KSEL + t) * 4;
        od[0] = nk ? box[t][0] : 0.f;
        od[1] = nk ? box[t][1] : 0.f;
        od[2] = nk ? box[t][2] : 0.f;
        od[3] = nk ? box[t][3] : 0.f;
    }
}

// ---------------------------------------------------------------------------
// Chars: bezier AABB via V_WMMA_F32_16X16X4_F32.
//   A (16x4): bezier basis, rows 0..9 = t=m/9, rows 10..15 replicate t=0
//             (duplicate sample points do not perturb min/max).
//   B (4x16): 4 candidates x {top,bottom} x {d0,d1} control-point columns.
//   D (16x16): 16 samples x 16 columns; reduce min/max per column, then
//              combine sample halves (shfl_xor 16) and curves (shfl_xor 2).
// One workgroup (4 waves) per image; wave w handles tiles w, w+4, ...
// ---------------------------------------------------------------------------
__global__ void char_filter_kernel(const int* __restrict__ idx_in,
                                   const float* __restrict__ val_in,
                                   const float* __restrict__ raw, int N,     // [B,N,16]
                                   const float* __restrict__ praw, int NP,   // [B,NP,4]
                                   const float* __restrict__ tsizes,
                                   float* __restrict__ out_data,
                                   float* __restrict__ out_scores,
                                   float* __restrict__ out_keep) {
    const int b = blockIdx.x;
    const int t = threadIdx.x;
    const int lane = t & 31;
    const int wid = t >> 5;
    const int* sidx = idx_in + b * KSEL;
    const float* vals = val_in + b * KSEL;
    const float imh = tsizes[2 * b + 0], imw = tsizes[2 * b + 1];

    __shared__ float box[KSEL][4];
    __shared__ float area[KSEL];
    __shared__ int keepf[KSEL];
    __shared__ float belong[KSEL];
    __shared__ int s_anyk, s_anyvk, s_fbi;

    // A-matrix per ISA layout: lane m (0..15): VGPR j = A[m][j] for K=0,1;
    // lanes 16..31: K=2,3.
    const int n = lane & 15;
    const int half = lane >> 4;
    v2f A;
    {
        float tt = (n <= 9) ? (float)n * (1.0f / 9.0f) : 0.0f;
        float ti = 1.0f - tt;
        float bas0 = ti * ti * ti;
        float bas1 = 3.f * tt * ti * ti;
        float bas2 = 3.f * tt * tt * ti;
        float bas3 = tt * tt * tt;
        A[0] = half ? bas2 : bas0;
        A[1] = half ? bas3 : bas1;
    }

    // WMMA phase: full EXEC per wave (tile loop bound uniform within wave).
    for (int tile = wid; tile < (KSEL / 4); tile += 4) {
        int cand = tile * 4 + (n >> 2);
        int curve = (n >> 1) & 1;
        int dim = n & 1;
        float sc = dim ? imw : imh;  // scale_pts = tile((h,w),8): d0*=h, d1*=w
        const float* rrow = raw + ((size_t)b * N + sidx[cand]) * 16;
        int fbase = half * 2;
        v2f Bm;
        Bm[0] = rrow[(curve * 4 + fbase + 0) * 2 + dim] * sc;
        Bm[1] = rrow[(curve * 4 + fbase + 1) * 2 + dim] * sc;
        v8f C = {0.f, 0.f, 0.f, 0.f, 0.f, 0.f, 0.f, 0.f};
        v8f D = __builtin_amdgcn_wmma_f32_16x16x4_f32(
            false, A, false, Bm, (short)0, C, false, false);
        float mn = D[0], mx = D[0];
#pragma unroll
        for (int r = 1; r < 8; ++r) { mn = fminf(mn, D[r]); mx = fmaxf(mx, D[r]); }
        mn = fminf(mn, __shfl_xor(mn, 16, 32));  // sample rows 8..15
        mx = fmaxf(mx, __shfl_xor(mx, 16, 32));
        mn = fminf(mn, __shfl_xor(mn, 2, 32));   // top/bottom curve
        mx = fmaxf(mx, __shfl_xor(mx, 2, 32));
        if (half == 0 && curve == 0) {
            box[cand][dim] = mn;       // [min_d0, min_d1, max_d0, max_d1]
            box[cand][2 + dim] = mx;
        }
    }
    __syncthreads();

    float v = 0.f;
    if (t < KSEL) {
        v = vals[t];
        keepf[t] = (v > SCORE_TH) ? 1 : 0;
        // parent = pred_line, ef = 1
        int pi = sidx[t];
        const float* pr = praw + ((size_t)b * NP + pi) * 4;
        float pcx = pr[0], pcy = pr[1], pw2 = pr[2] * 0.5f, ph2 = pr[3] * 0.5f;
        float px1 = (pcx - pw2) * imw, py1 = (pcy - ph2) * imh;
        float px2 = (pcx + pw2) * imw, py2 = (pcy + ph2) * imh;
        float x1 = box[t][0], y1 = box[t][1], x2 = box[t][2], y2 = box[t][3];
        float ix1 = fmaxf(x1, px1), iy1 = fmaxf(y1, py1);
        float ix2 = fminf(x2, px2), iy2 = fminf(y2, py2);
        float inter = fmaxf(ix2 - ix1, 0.f) * fmaxf(iy2 - iy1, 0.f);
        float ca = (x2 - x1) * (y2 - y1);
        belong[t] = inter / (ca + 1e-6f);
    }
    __syncthreads();
    if (t == 0) {
        int a = 0;
        for (int j = 0; j < KSEL; ++j) a |= keepf[j];
        s_anyk = a;
    }
    __syncthreads();
    if (t < KSEL) keepf[t] = s_anyk ? keepf[t] : (t == 0);
    __syncthreads();
    if (t == 0) {
        int avk = 0; float fv = -INFINITY; int fi = 0;
        for (int j = 0; j < KSEL; ++j) {
            if (keepf[j] && belong[j] > BELONG_TH) avk = 1;
            float s = keepf[j] ? belong[j] : -INFINITY;
            if (s > fv) { fv = s; fi = j; }
        }
        s_anyvk = avk; s_fbi = fi;
    }
    __syncthreads();
    if (t < KSEL) {
        int valid = s_anyvk ? (belong[t] > BELONG_TH) : (t == s_fbi);
        keepf[t] = keepf[t] && valid;
    }
    __syncthreads();
    nms_inplace(box, area, keepf, t);
    if (t < KSEL) {
        int nk = keepf[t];
        out_scores[b * KSEL + t] = nk ? v : 0.f;
        out_keep[b * KSEL + t] = nk ? 1.f : 0.f;
    }
    __syncthreads();
    // data = scaled control points (16 floats per candidate), zeroed if !keep
    for (int i = t; i < KSEL * 16; i += blockDim.x) {
        int c = i >> 4, p = i & 15;
        float val = 0.f;
        if (keepf[c]) {
            val = raw[((size_t)b * N + sidx[c]) * 16 + p] * ((p & 1) ? imw : imh);
        }
        out_data[((size_t)b * KSEL + c) * 16 + p] = val;
    }
}

// ---------------------------------------------------------------------------
extern "C" void kernel_launch(void* const* d_in, const int* in_sizes, int n_in,
                              void* d_out, int out_size, void* d_ws, size_t ws_size,
                              hipStream_t stream) {
    (void)n_in; (void)out_size; (void)ws_size;
    const float* blk_lg = (const float*)d_in[0];
    const float* lin_lg = (const float*)d_in[1];
    const float* chr_lg = (const float*)d_in[2];
    const float* blk_raw = (const float*)d_in[3];
    const float* lin_raw = (const float*)d_in[4];
    const float* chr_raw = (const float*)d_in[5];
    const float* tsz = (const float*)d_in[6];

    const int B = in_sizes[6] / 2;
    const int NB = in_sizes[0] / B;
    const int NL = in_sizes[1] / B;
    const int NC = in_sizes[2] / B;

    int* ws_idx = (int*)d_ws;                          // 3*B*KSEL ints
    float* ws_val = (float*)(ws_idx + 3 * B * KSEL);   // 3*B*KSEL floats

    topk_sigmoid_kernel<<<B, 256, 0, stream>>>(blk_lg, NB, ws_idx, ws_val);
    topk_sigmoid_kernel<<<B, 256, 0, stream>>>(lin_lg, NL, ws_idx + B * KSEL,
                                               ws_val + B * KSEL);
    topk_sigmoid_kernel<<<B, 256, 0, stream>>>(chr_lg, NC, ws_idx + 2 * B * KSEL,
                                               ws_val + 2 * B * KSEL);

    float* out = (float*)d_out;
    float* blk_data = out;
    float* blk_sc = blk_data + (size_t)B * KSEL * 4;
    float* blk_kp = blk_sc + (size_t)B * KSEL;
    float* lin_data = blk_kp + (size_t)B * KSEL;
    float* lin_sc = lin_data + (size_t)B * KSEL * 4;
    float* lin_kp = lin_sc + (size_t)B * KSEL;
    float* chr_data = lin_kp + (size_t)B * KSEL;
    float* chr_sc = chr_data + (size_t)B * KSEL * 16;
    float* chr_kp = chr_sc + (size_t)B * KSEL;

    box_filter_kernel<<<B, 128, 0, stream>>>(ws_idx, ws_val, blk_raw, NB,
                                             nullptr, 0, 1, tsz,
                                             blk_data, blk_sc, blk_kp);
    box_filter_kernel<<<B, 128, 0, stream>>>(ws_idx + B * KSEL, ws_val + B * KSEL,
                                             lin_raw, NL, blk_raw, NB, 4, tsz,
                                             lin_data, lin_sc, lin_kp);
    char_filter_kernel<<<B, 128, 0, stream>>>(ws_idx + 2 * B * KSEL,
                                              ws_val + 2 * B * KSEL,
                                              chr_raw, NC, lin_raw, NL, tsz,
                                              chr_data, chr_sc, chr_kp);
}